// ClippedReLU_45406394253468
// MI455X (gfx1250) — compile-verified
//
#include <hip/hip_runtime.h>
#include <stdint.h>

#define TPB    256
#define WAVES  (TPB / 32)
#define DEPTH  8
#define MCHUNK 64
#define FDIM   1024

// ---------------------------------------------------------------------------
// Native clang vector types (HIP_vector_type classes are rejected by
// __builtin_nontemporal_store and the async-LDS builtins).
// ---------------------------------------------------------------------------
typedef float v4f __attribute__((ext_vector_type(4)));
typedef int   v4i __attribute__((ext_vector_type(4)));
typedef __attribute__((address_space(1))) v4i gv4i_t;   // global int4
typedef __attribute__((address_space(3))) v4i lv4i_t;   // LDS int4

// ---------------------------------------------------------------------------
// CDNA5 async global->LDS copy (tracked by ASYNCcnt) + wait helpers.
// ---------------------------------------------------------------------------
__device__ __forceinline__ void async_copy_b128(const float* gsrc, void* lds_dst) {
#if __has_builtin(__builtin_amdgcn_global_load_async_to_lds_b128)
  __builtin_amdgcn_global_load_async_to_lds_b128(
      (gv4i_t*)(uintptr_t)gsrc,
      (lv4i_t*)(uint32_t)(uintptr_t)lds_dst,   // low 32 bits = LDS byte offset
      0, 0);
#else
  uint32_t lds_off = (uint32_t)(uintptr_t)lds_dst;
  asm volatile("global_load_async_to_lds_b128 %0, %1, off"
               :: "v"(lds_off), "v"(gsrc) : "memory");
#endif
}

#if __has_builtin(__builtin_amdgcn_s_wait_asynccnt)
#define WAIT_ASYNC(n) __builtin_amdgcn_s_wait_asynccnt(n)
#else
#define WAIT_ASYNC(n) asm volatile("s_wait_asynccnt " #n ::: "memory")
#endif

// piecewise-linear: y0 below x0, y1 above x1, linear ramp between
__device__ __forceinline__ float pwl1(float zz, float y0, float y1,
                                      float x0, float x1, float sl) {
  float lin = __builtin_fmaf(sl, zz - x0, y0);
  float hi  = (zz <= x1) ? lin : y1;
  return (zz < x0) ? y0 : hi;
}

__global__ void __launch_bounds__(TPB)
clipped_pwl_kernel(const float* __restrict__ z, const int* __restrict__ mask,
                   const float* __restrict__ eta, float* __restrict__ out,
                   int Ndim, int Mdim) {
  // per-wave depth-8 staging buffers: 8 waves x 8 x 512B = 32 KB
  __shared__ v4f stage[WAVES][DEPTH][32];

  const int t    = threadIdx.x;
  const int wave = t >> 5;
  const int lane = t & 31;
  const int b    = blockIdx.z;
  const int n    = blockIdx.y;
  const int m0   = blockIdx.x * MCHUNK;
  const int f0   = t << 2;                 // this thread's fixed f-quad

  // ---- per-thread parameters (loaded once, held in VGPRs) ----
  const v4i mrow = *(const v4i*)(mask + b * FDIM + f0);
  const int mr[4] = {mrow.x, mrow.y, mrow.z, mrow.w};
  float y0[4], y1[4], x0[4], x1[4], sl[4];
#pragma unroll
  for (int j = 0; j < 4; ++j) {
    const v4f e = *(const v4f*)(eta + mr[j] * 4);
    y0[j] = e.x; y1[j] = e.y; x0[j] = e.z; x1[j] = e.w;
    sl[j] = (y1[j] - y0[j]) / (x1[j] - x0[j]);
  }

  const size_t rowStride = (size_t)FDIM;
  const size_t base = (((size_t)b * Ndim + n) * Mdim + m0) * rowStride + (size_t)f0;
  const float* gsrc = z + base;
  float*       gdst = out + base;

  // ---- prologue: fill the async pipeline (8 rows in flight) ----
#pragma unroll
  for (int d = 0; d < DEPTH; ++d)
    async_copy_b128(gsrc + (size_t)d * rowStride, &stage[wave][d][lane]);

  auto process = [&](int m) {
    v4f v = stage[wave][m & (DEPTH - 1)][lane];   // ds_load_b128
    v4f r;
    r.x = pwl1(v.x, y0[0], y1[0], x0[0], x1[0], sl[0]);
    r.y = pwl1(v.y, y0[1], y1[1], x0[1], x1[1], sl[1]);
    r.z = pwl1(v.z, y0[2], y1[2], x0[2], x1[2], sl[2]);
    r.w = pwl1(v.w, y0[3], y1[3], x0[3], x1[3], sl[3]);
    __builtin_nontemporal_store(r, (v4f*)(gdst + (size_t)m * rowStride));
  };

  // ---- steady state: wait oldest, consume, refill same buffer ----
#pragma unroll 8
  for (int m = 0; m < MCHUNK - DEPTH; ++m) {
    WAIT_ASYNC(7);                 // oldest of 8 in-flight rows has landed
    process(m);                    // data use forces DScnt wait before reuse
    async_copy_b128(gsrc + (size_t)(m + DEPTH) * rowStride,
                    &stage[wave][m & (DEPTH - 1)][lane]);
  }

  // ---- epilogue: drain pipeline (counter counts instructions, in order) ----
  WAIT_ASYNC(7); process(MCHUNK - 8);
  WAIT_ASYNC(6); process(MCHUNK - 7);
  WAIT_ASYNC(5); process(MCHUNK - 6);
  WAIT_ASYNC(4); process(MCHUNK - 5);
  WAIT_ASYNC(3); process(MCHUNK - 4);
  WAIT_ASYNC(2); process(MCHUNK - 3);
  WAIT_ASYNC(1); process(MCHUNK - 2);
  WAIT_ASYNC(0); process(MCHUNK - 1);
}

extern "C" void kernel_launch(void* const* d_in, const int* in_sizes, int n_in,
                              void* d_out, int out_size, void* d_ws, size_t ws_size,
                              hipStream_t stream) {
  const float* z    = (const float*)d_in[0];
  const int*   mask = (const int*)d_in[1];
  const float* eta  = (const float*)d_in[2];
  float*       out  = (float*)d_out;

  const int B    = in_sizes[1] / FDIM;                                  // 4
  const long long total = (long long)in_sizes[0];                       // B*N*M*F
  const int NM   = (int)(total / ((long long)B * FDIM));                // N*M
  const int Mdim = 1024;
  const int Ndim = NM / Mdim;                                           // 16

  dim3 grid(Mdim / MCHUNK, Ndim, B);
  clipped_pwl_kernel<<<grid, TPB, 0, stream>>>(z, mask, eta, out, Ndim, Mdim);
}